// HybridReIDModel_39084202394053
// MI455X (gfx1250) — compile-verified
//
#include <hip/hip_runtime.h>
#include <math.h>

typedef __attribute__((ext_vector_type(2))) float v2f;
typedef __attribute__((ext_vector_type(8))) float v8f;

#define N_Q   6
#define N_CL  32
#define N_MOD 32
#define PART  64
#define DIM   64      // 2^N_Q
#define NCLS  150
#define KFEAT (N_MOD * N_Q)   // 192

// ---------------------------------------------------------------------------
// Kernel 1: angles = parts @ Wp^T + bp ; build initial product state
//   S[p][j][b] = prod_q ( bit_q(j) ? sin(angle/2) : cos(angle/2) )
// qubit q <-> bit position (5-q) of flattened index j.
// ---------------------------------------------------------------------------
__global__ void init_state_kernel(const float* __restrict__ features, // [B][2048]
                                  const float* __restrict__ Wp,       // [6][64]
                                  const float* __restrict__ bp,       // [6]
                                  float* __restrict__ S,              // [32][64][B]
                                  int B)
{
    int b = blockIdx.x * blockDim.x + threadIdx.x;
    int p = blockIdx.y;
    if (b >= B) return;

    const float* f = features + (size_t)b * 2048 + p * PART;

    float ang[N_Q];
#pragma unroll
    for (int q = 0; q < N_Q; ++q) ang[q] = bp[q];

    for (int d = 0; d < PART; ++d) {
        float fd = f[d];
#pragma unroll
        for (int q = 0; q < N_Q; ++q) ang[q] = fmaf(fd, Wp[q * PART + d], ang[q]);
    }

    float c[N_Q], s[N_Q];
#pragma unroll
    for (int q = 0; q < N_Q; ++q) {
        float h = ang[q] * 0.5f;
        c[q] = __cosf(h);
        s[q] = __sinf(h);
    }

    float* Sp = S + ((size_t)p * DIM) * B + b;
#pragma unroll
    for (int j = 0; j < DIM; ++j) {
        float v = 1.0f;
#pragma unroll
        for (int q = 0; q < N_Q; ++q)
            v *= ((j >> (5 - q)) & 1) ? s[q] : c[q];
        Sp[(size_t)j * B] = v;
    }
}

// ---------------------------------------------------------------------------
// Kernel 2: build the total 64x64 complex unitary per module by pushing the
// 64 basis columns through 32 layers of (6 RX gates + CNOT ring) in LDS.
// One workgroup (256 threads) per module.
//   U layout: U[p][0][r][c] = real, U[p][1][r][c] = imag   (r,c in [0,64))
// ---------------------------------------------------------------------------
__global__ void build_unitary_kernel(const float* __restrict__ qw, // [32][32][6]
                                     float* __restrict__ U)        // [32][2][64][64]
{
    __shared__ float ur[DIM][DIM];
    __shared__ float ui[DIM][DIM];

    int p = blockIdx.x;
    int t = threadIdx.x;

    // identity
    for (int idx = t; idx < DIM * DIM; idx += 256) {
        int r = idx >> 6, cidx = idx & 63;
        ur[r][cidx] = (r == cidx) ? 1.0f : 0.0f;
        ui[r][cidx] = 0.0f;
    }
    __syncthreads();

    for (int l = 0; l < N_CL; ++l) {
        // --- 6 RX gates ---
        for (int q = 0; q < N_Q; ++q) {
            float th = qw[((size_t)p * N_CL + l) * N_Q + q] * 0.5f;
            float cg = __cosf(th);
            float sg = __sinf(th);
            int bp_ = 5 - q;                 // bit position for qubit q
            // 32 row-pairs x 64 columns = 2048 work items
            for (int i = t; i < 32 * DIM; i += 256) {
                int col = i & 63;
                int pr  = i >> 6;            // 0..31
                int low = pr & ((1 << bp_) - 1);
                int r0  = ((pr - low) << 1) | low;     // insert 0 at bit bp_
                int r1  = r0 | (1 << bp_);
                float a0r = ur[r0][col], a0i = ui[r0][col];
                float a1r = ur[r1][col], a1i = ui[r1][col];
                // RX: new0 = c*a0 - i*s*a1 ; new1 = -i*s*a0 + c*a1
                ur[r0][col] = cg * a0r + sg * a1i;
                ui[r0][col] = cg * a0i - sg * a1r;
                ur[r1][col] = cg * a1r + sg * a0i;
                ui[r1][col] = cg * a1i - sg * a0r;
            }
            __syncthreads();
        }
        // --- CNOT ring: ctrl q -> tgt (q+1)%6 ---
        for (int q = 0; q < N_Q; ++q) {
            int cb = 5 - q;
            int tb = 5 - ((q + 1) % N_Q);
            // 16 row-pairs (ctrl=1, tgt=0) x 64 columns
            for (int i = t; i < 16 * DIM; i += 256) {
                int col = i & 63;
                int pr  = i >> 6;            // 0..15 over free bits
                int r0 = 0, k = 0;
#pragma unroll
                for (int bpos = 0; bpos < 6; ++bpos) {
                    if (bpos == cb)      r0 |= (1 << bpos);
                    else if (bpos == tb) { /* tgt bit = 0 */ }
                    else                 { r0 |= ((pr >> k) & 1) << bpos; ++k; }
                }
                int r1 = r0 | (1 << tb);
                float tr = ur[r0][col]; ur[r0][col] = ur[r1][col]; ur[r1][col] = tr;
                float ti = ui[r0][col]; ui[r0][col] = ui[r1][col]; ui[r1][col] = ti;
            }
            __syncthreads();
        }
    }

    float* Up = U + (size_t)p * 2 * DIM * DIM;
    for (int idx = t; idx < DIM * DIM; idx += 256) {
        Up[idx]              = ur[idx >> 6][idx & 63];
        Up[DIM * DIM + idx]  = ui[idx >> 6][idx & 63];
    }
}

// ---------------------------------------------------------------------------
// Kernel 3: batched WMMA GEMM.  Per module p, per 16-batch tile:
//   Yr = Ur^p (64x64) @ S_p (64x16),  Yi = Ui^p @ S_p
//   probs = Yr^2 + Yi^2 ;  outs[b][p*6+q] = sum_j sign(j,q) * probs[j][b]
// One wave (32 threads) per block; 128 v_wmma_f32_16x16x4_f32 per wave.
// ---------------------------------------------------------------------------
__global__ void __launch_bounds__(32)
quantum_apply_kernel(const float* __restrict__ S,   // [32][64][B]
                     const float* __restrict__ U,   // [32][2][64][64]
                     float* __restrict__ outs,      // [B][192]
                     int B)
{
    int p    = blockIdx.y;
    int b0   = blockIdx.x * 16;
    int lane = threadIdx.x;
    int half = lane >> 4;      // 0 or 1
    int lm   = lane & 15;

    const float* Ur = U + (size_t)p * 2 * DIM * DIM;
    const float* Ui = Ur + DIM * DIM;
    const float* Sp = S + (size_t)p * DIM * B;

    v8f accr[4], acci[4];
#pragma unroll
    for (int m = 0; m < 4; ++m) {
        accr[m] = (v8f)(0.0f);
        acci[m] = (v8f)(0.0f);
    }

    for (int k = 0; k < 16; ++k) {           // K = 64 in steps of 4
        int kb = k * 4 + 2 * half;           // this lane-half's K offset
        v2f bm;
        bm.x = Sp[(size_t)(kb + 0) * B + b0 + lm];
        bm.y = Sp[(size_t)(kb + 1) * B + b0 + lm];
#pragma unroll
        for (int m = 0; m < 4; ++m) {
            int row = m * 16 + lm;
            v2f ar, ai;
            ar.x = Ur[row * DIM + kb + 0];
            ar.y = Ur[row * DIM + kb + 1];
            ai.x = Ui[row * DIM + kb + 0];
            ai.y = Ui[row * DIM + kb + 1];
            accr[m] = __builtin_amdgcn_wmma_f32_16x16x4_f32(
                false, ar, false, bm, (short)0, accr[m], false, false);
            acci[m] = __builtin_amdgcn_wmma_f32_16x16x4_f32(
                false, ai, false, bm, (short)0, acci[m], false, false);
        }
    }

    // probs + sign fold.  C/D layout: VGPR r -> M = r + 8*half ; lane -> N = lm.
    float o[N_Q];
#pragma unroll
    for (int q = 0; q < N_Q; ++q) o[q] = 0.0f;

#pragma unroll
    for (int m = 0; m < 4; ++m) {
#pragma unroll
        for (int r = 0; r < 8; ++r) {
            int j = m * 16 + r + 8 * half;
            float yr = accr[m][r], yi = acci[m][r];
            float pr = yr * yr + yi * yi;
#pragma unroll
            for (int q = 0; q < N_Q; ++q)
                o[q] += ((j >> (5 - q)) & 1) ? -pr : pr;
        }
    }

    // combine the two lane halves (each half covered a disjoint j set, same N)
#pragma unroll
    for (int q = 0; q < N_Q; ++q)
        o[q] += __shfl_xor(o[q], 16, 32);

    if (lane < 16) {
#pragma unroll
        for (int q = 0; q < N_Q; ++q)
            outs[(size_t)(b0 + lane) * KFEAT + p * N_Q + q] = o[q];
    }
}

// ---------------------------------------------------------------------------
// Kernel 4: classifier GEMM  out[B][150] = outs[B][192] @ Wf^T + bf
// WMMA f32 16x16x4; grid (B/16, 10) covers 160 class columns, guard to 150.
// Branchless clamped loads keep EXEC full around WMMA.
// ---------------------------------------------------------------------------
__global__ void __launch_bounds__(32)
final_gemm_kernel(const float* __restrict__ X,   // [B][192]
                  const float* __restrict__ Wf,  // [150][192]
                  const float* __restrict__ bf,  // [150]
                  float* __restrict__ out,       // [B][150]
                  int B)
{
    int b0   = blockIdx.x * 16;
    int n0   = blockIdx.y * 16;
    int lane = threadIdx.x;
    int half = lane >> 4;
    int lm   = lane & 15;

    int n      = n0 + lm;
    float nsel = (n < NCLS) ? 1.0f : 0.0f;
    int nc     = (n < NCLS) ? n : (NCLS - 1);

    v8f acc = (v8f)(0.0f);

    for (int k = 0; k < KFEAT / 4; ++k) {    // 48 K-steps
        int kb = k * 4 + 2 * half;
        v2f a, bm;
        a.x  = X[(size_t)(b0 + lm) * KFEAT + kb + 0];
        a.y  = X[(size_t)(b0 + lm) * KFEAT + kb + 1];
        bm.x = Wf[(size_t)nc * KFEAT + kb + 0] * nsel;
        bm.y = Wf[(size_t)nc * KFEAT + kb + 1] * nsel;
        acc = __builtin_amdgcn_wmma_f32_16x16x4_f32(
            false, a, false, bm, (short)0, acc, false, false);
    }

    if (n < NCLS) {
        float bias = bf[n];
#pragma unroll
        for (int r = 0; r < 8; ++r) {
            int brow = b0 + r + 8 * half;
            out[(size_t)brow * NCLS + n] = acc[r] + bias;
        }
    }
}

// ---------------------------------------------------------------------------
extern "C" void kernel_launch(void* const* d_in, const int* in_sizes, int n_in,
                              void* d_out, int out_size, void* d_ws, size_t ws_size,
                              hipStream_t stream) {
    const float* features = (const float*)d_in[0];  // [B][2048]
    const float* Wp       = (const float*)d_in[1];  // [6][64]
    const float* bp       = (const float*)d_in[2];  // [6]
    const float* qw       = (const float*)d_in[3];  // [32][32][6]
    const float* Wf       = (const float*)d_in[4];  // [150][192]
    const float* bf       = (const float*)d_in[5];  // [150]
    float* out            = (float*)d_out;          // [B][150]

    const int B = in_sizes[0] / 2048;               // 2048

    // workspace layout
    float* S  = (float*)d_ws;                       // 32*64*B floats   (16 MB)
    float* U  = S + (size_t)N_MOD * DIM * B;        // 32*2*64*64       ( 1 MB)
    float* XO = U + (size_t)N_MOD * 2 * DIM * DIM;  // B*192            (1.5 MB)

    // 1) angles + initial product states
    {
        dim3 grid((B + 255) / 256, N_MOD);
        init_state_kernel<<<grid, 256, 0, stream>>>(features, Wp, bp, S, B);
    }
    // 2) per-module total unitaries
    build_unitary_kernel<<<N_MOD, 256, 0, stream>>>(qw, U);
    // 3) batched WMMA state evolution + expectation fold
    {
        dim3 grid(B / 16, N_MOD);
        quantum_apply_kernel<<<grid, 32, 0, stream>>>(S, U, XO, B);
    }
    // 4) classifier WMMA GEMM
    {
        dim3 grid(B / 16, (NCLS + 15) / 16);
        final_gemm_kernel<<<grid, 32, 0, stream>>>(XO, Wf, bf, out, B);
    }
}